// IncConvModule_54606214201886
// MI455X (gfx1250) — compile-verified
//
#include <hip/hip_runtime.h>

// ---------------------------------------------------------------------------
// IncConv patch update for MI455X (gfx1250, wave32).
//
// Cost model: output copy (411 MB @ 23.3 TB/s ~ 18us) dominates; the patch
// conv is ~1.2 GFLOP -> implicit GEMM on the fp32 WMMA path
// (v_wmma_f32_16x16x4_f32), M=Cout=64, N=32*32=1024, K=Cin*9=576.
// K is enumerated (kh,kw)-outer x ci-inner: boundary masks and address math
// hoist out of the branchless, division-free inner loop (EXEC stays all-1s).
// Weights are pre-transposed to (khw, ci, co) in d_ws so A-fragment loads are
// lane-coalesced (one cache line per load) like the B loads already are.
// The bulk copy streams with non-temporal hints so the 411MB pass doesn't
// evict the L2-resident conv operands (weights 147KB + input slabs).
// ---------------------------------------------------------------------------

typedef __attribute__((ext_vector_type(2))) float v2f;
typedef __attribute__((ext_vector_type(4))) float v4f;   // native vector: OK for NT builtins
typedef __attribute__((ext_vector_type(8))) float v8f;

#define B_    16
#define CIN   64
#define H_    224
#define W_    224
#define HW_   (H_ * W_)
#define COUT  64
#define PH_   32
#define KDIM  (CIN * 9)             // 576
#define WT_ELEMS (9 * CIN * COUT)   // 36864 floats = 147KB

// ---------------------------------------------------------------------------
// Kernel 0: one-shot weight transpose (Cout,Cin,3,3) -> (khw,ci,co).
// ---------------------------------------------------------------------------
__global__ void transpose_w(const float* __restrict__ src, float* __restrict__ dst) {
  int idx = blockIdx.x * blockDim.x + threadIdx.x;
  if (idx >= WT_ELEMS) return;
  int co  = idx % COUT;
  int t   = idx / COUT;
  int ci  = t % CIN;
  int khw = t / CIN;
  dst[idx] = src[(size_t)co * KDIM + ci * 9 + khw];   // coalesced store
}

// ---------------------------------------------------------------------------
// Kernel 1: bulk copy of the stale output buffer, b128 vectorized, NT hints.
// ---------------------------------------------------------------------------
__global__ void copy_out_b128(const v4f* __restrict__ src,
                              v4f* __restrict__ dst,
                              int n4) {
  int idx = blockIdx.x * blockDim.x + threadIdx.x;
  int stride = gridDim.x * blockDim.x;
  for (int i = idx; i < n4; i += stride) {
    v4f v = __builtin_nontemporal_load(&src[i]);
    __builtin_nontemporal_store(v, &dst[i]);
  }
}

// ---------------------------------------------------------------------------
// Kernel 2: patch conv as implicit GEMM with V_WMMA_F32_16X16X4_F32.
// One block (8 wave32) per (batch, patch-row): 4 M-tiles x 2 N-tiles.
// Per wave: 9 outer (kh,kw) x 16 inner (ci/4) = 144 WMMAs.
// WT=true : w is transposed (khw,ci,co)   -> coalesced A loads (preferred)
// WT=false: w is original (co,ci,kh,kw)   -> strided A loads (ws fallback)
// ---------------------------------------------------------------------------
template <bool WT>
__global__ void __launch_bounds__(256)
patch_conv_wmma(const float* __restrict__ x,      // (B, Cin, H, W)
                const float* __restrict__ w,      // layout per WT
                const float* __restrict__ bias,   // (Cout,)
                const int*   __restrict__ loc,    // (B, 2) = (y, x)
                float* __restrict__ out)          // (B, Cout, H, W)
{
  const int blk    = blockIdx.x;
  const int b      = blk >> 5;        // batch
  const int py     = blk & 31;        // patch output row (block-uniform)
  const int wave   = threadIdx.x >> 5;
  const int lane   = threadIdx.x & 31;
  const int tileM  = wave & 3;        // which 16 Cout rows
  const int nhalf  = wave >> 2;       // which 16 patch columns
  const int lane16 = lane & 15;
  const int hi     = lane >> 4;       // K-half select for A/B fragments

  const int loc_y = loc[b * 2 + 0];
  const int loc_x = loc[b * 2 + 1];

  const int px = nhalf * 16 + lane16;      // patch col 0..31 (N index)
  const int m  = tileM * 16 + lane16;      // A-matrix row (Cout) for this lane

  const float* __restrict__ wbase = WT ? (w + m) : (w + (size_t)m * KDIM);
  const float* __restrict__ xb    = x + (size_t)b * CIN * HW_;

  v8f acc = {};

  // K enumeration: k_lin = (kh*3+kw)*CIN + ci  (A and B agree; sum order free)
  for (int kh = 0; kh < 3; ++kh) {
    const int iy  = loc_y + py + kh - 1;           // padding = 1
    const int iyc = iy < 0 ? 0 : (iy > H_ - 1 ? H_ - 1 : iy);
    const bool yok = (unsigned)iy < (unsigned)H_;
    for (int kw = 0; kw < 3; ++kw) {
      const int khw = kh * 3 + kw;
      const int ix  = loc_x + px + kw - 1;
      const int ixc = ix < 0 ? 0 : (ix > W_ - 1 ? W_ - 1 : ix);
      const float msk = (yok && (unsigned)ix < (unsigned)W_) ? 1.0f : 0.0f;

      const float* __restrict__ xoff = xb + (size_t)iyc * W_ + ixc;   // + ci*HW
      const float* __restrict__ woff =
          WT ? (wbase + (size_t)khw * CIN * COUT)                     // + ci*COUT
             : (wbase + khw);                                         // + ci*9

#pragma unroll 4
      for (int ci = 0; ci < CIN; ci += 4) {
        const int c0 = ci + 2 * hi;   // this lane's K pair: ci=c0, c0+1
        v2f a;
        if (WT) {
          a.x = woff[(size_t)c0 * COUT];          // lanes 0-15: contiguous 64B
          a.y = woff[(size_t)(c0 + 1) * COUT];
        } else {
          a.x = woff[c0 * 9];
          a.y = woff[(c0 + 1) * 9];
        }
        v2f bf;
        bf.x = xoff[(size_t)c0 * HW_] * msk;      // branchless zero-pad
        bf.y = xoff[(size_t)(c0 + 1) * HW_] * msk;
        acc = __builtin_amdgcn_wmma_f32_16x16x4_f32(
            /*neg_a=*/false, a, /*neg_b=*/false, bf,
            /*c_mod=*/(short)0, acc, /*reuse_a=*/false, /*reuse_b=*/false);
      }
    }
  }

  // --- C/D layout: VGPR v -> M = tileM*16 + v + 8*hi, N = lane16
  const int oy = loc_y + py;
  const int ox = loc_x + px;
#pragma unroll
  for (int v = 0; v < 8; ++v) {
    const int co = tileM * 16 + v + 8 * hi;
    const float r = acc[v] + bias[co];
    out[((size_t)(b * COUT + co) * H_ + oy) * W_ + ox] = r;
  }
}

// ---------------------------------------------------------------------------
// Launch
// ---------------------------------------------------------------------------
extern "C" void kernel_launch(void* const* d_in, const int* in_sizes, int n_in,
                              void* d_out, int out_size, void* d_ws, size_t ws_size,
                              hipStream_t stream) {
  const float* x    = (const float*)d_in[0];   // in_tensor
  const float* w    = (const float*)d_in[1];   // weights
  const float* bias = (const float*)d_in[2];   // biases
  const float* oin  = (const float*)d_in[3];   // stale out_tensor
  const int*   loc  = (const int*)d_in[4];     // patch locations (B,2)
  float* out = (float*)d_out;

  // 1) copy stale output -> d_out (HBM-bound part, NT b128 streaming)
  const int n4 = out_size / 4;                 // out_size divisible by 4
  copy_out_b128<<<8192, 256, 0, stream>>>((const v4f*)oin, (v4f*)out, n4);

  // 2) overwrite the 16 patches via WMMA implicit GEMM (same stream -> ordered)
  if (ws_size >= (size_t)WT_ELEMS * sizeof(float)) {
    float* wT = (float*)d_ws;
    transpose_w<<<(WT_ELEMS + 255) / 256, 256, 0, stream>>>(w, wT);
    patch_conv_wmma<true><<<B_ * PH_, 256, 0, stream>>>(x, wT, bias, loc, out);
  } else {
    patch_conv_wmma<false><<<B_ * PH_, 256, 0, stream>>>(x, w, bias, loc, out);
  }
}